// ClusterGCN_58506044506346
// MI455X (gfx1250) — compile-verified
//
#include <hip/hip_runtime.h>

typedef __attribute__((ext_vector_type(2))) float v2f;
typedef __attribute__((ext_vector_type(8))) float v8f;

#define NNODES 50000
#define NEDGES 800000
#define FEAT   128
#define LDSTR  132   // LDS row stride: 132%64=4 -> 16 consecutive rows land on distinct banks

static __device__ __forceinline__ int imin(int a, int b) { return a < b ? a : b; }

// ---------------- zero (float4) ----------------
__global__ void zero_f32v4(float4* __restrict__ p, int n4) {
    int i = blockIdx.x * blockDim.x + threadIdx.x;
    if (i < n4) p[i] = make_float4(0.f, 0.f, 0.f, 0.f);
}

// ---------------- degree (layer-invariant) ----------------
__global__ void degree_kernel(const int* __restrict__ dst, float* __restrict__ deg, int E) {
    int e = blockIdx.x * blockDim.x + threadIdx.x;
    if (e < E) atomicAdd(&deg[dst[e]], 1.0f);
}

// ---------------- edge scatter: summed[dst] += x[src]  (one wave per edge) ----------------
__global__ void scatter_kernel(const float* __restrict__ xin,
                               const int* __restrict__ src,
                               const int* __restrict__ dst,
                               float* __restrict__ summed, int E) {
    int t    = blockIdx.x * blockDim.x + threadIdx.x;
    int e    = t >> 5;          // one wave per edge
    int lane = t & 31;          // each lane: 4 consecutive floats (512B/wave, coalesced)
    if (e >= E) return;
    int s = src[e];
    int d = dst[e];
    const float4 v = ((const float4*)(xin + (long)s * FEAT))[lane];
    float* o = summed + (long)d * FEAT + lane * 4;
    atomicAdd(o + 0, v.x);
    atomicAdd(o + 1, v.y);
    atomicAdd(o + 2, v.z);
    atomicAdd(o + 3, v.w);
}

// ---------------- fused SAGE layer: out = relu?( mean@Wl + bl + x@Wr ) ----------------
// Block: 256 threads = 8 waves, owns 64 rows (4 row tiles of 16).
// Wave w owns column tile(s) c = w, w+8, ...; each B fragment is reused across the
// 4 row tiles (4 independent accumulators -> 4-deep WMMA interleave, no D->A/B hazard).
// Out-of-range columns (DOUT=47 tail) are handled by clamping the load column; the
// garbage lands only in output columns that are never stored. No divergent loads.
template <int DOUT, bool RELU>
__global__ void __launch_bounds__(256)
sage_wmma(const float* __restrict__ xin, const float* __restrict__ summed,
          const float* __restrict__ deg, const float* __restrict__ Wl,
          const float* __restrict__ bl,  const float* __restrict__ Wr,
          float* __restrict__ out, int nrows) {
    constexpr int ROWS = 64;
    constexpr int NT   = (DOUT + 15) / 16;   // column tiles

    __shared__ float sMean[ROWS * LDSTR];
    __shared__ float sX[ROWS * LDSTR];
    __shared__ float sInv[ROWS];

    const int tid  = threadIdx.x;
    const int row0 = blockIdx.x * ROWS;

    // 1/max(deg,1) per row (clamped row index; padded rows never stored)
    for (int r = tid; r < ROWS; r += 256) {
        float d = deg[imin(row0 + r, nrows - 1)];
        sInv[r] = 1.0f / fmaxf(d, 1.0f);
    }
    __syncthreads();

    // stage x tile and mean tile into LDS, float4 at a time (branch-free)
    for (int idx = tid; idx < ROWS * (FEAT / 4); idx += 256) {
        int r  = idx >> 5;          // FEAT/4 = 32 float4 per row
        int k4 = (idx & 31) << 2;
        int g  = imin(row0 + r, nrows - 1);
        const float4 xv = *(const float4*)(xin    + (long)g * FEAT + k4);
        const float4 sv = *(const float4*)(summed + (long)g * FEAT + k4);
        float inv = sInv[r];
        *(float4*)(sX + r * LDSTR + k4) = xv;
        *(float4*)(sMean + r * LDSTR + k4) =
            make_float4(sv.x * inv, sv.y * inv, sv.z * inv, sv.w * inv);
    }
    __syncthreads();

    const int wave  = tid >> 5;
    const int lane  = tid & 31;
    const int mrow  = lane & 15;        // A row owned by this lane (within a 16-row tile)
    const int khalf = (lane >> 4) * 2;  // lanes 0-15: K pair {0,1}; lanes 16-31: {2,3}
    const int moff  = (lane >> 4) * 8;  // C/D: lanes 16-31 hold M = r + 8

    for (int c = wave; c < NT; c += 8) {
        const int n  = c * 16 + mrow;               // output column
        const int nl = imin(n, DOUT - 1);           // clamped load column (in-bounds)

        v8f acc[4];
        #pragma unroll
        for (int rt = 0; rt < 4; ++rt)
            acc[rt] = (v8f){0.f, 0.f, 0.f, 0.f, 0.f, 0.f, 0.f, 0.f};

        const float* aM = sMean + mrow * LDSTR + khalf;  // 8B-aligned -> ds_load_b64
        const float* aX = sX    + mrow * LDSTR + khalf;
        const float* wl = Wl + khalf * DOUT + nl;
        const float* wr = Wr + khalf * DOUT + nl;

        // mean @ Wl
        #pragma unroll 4
        for (int k0 = 0; k0 < FEAT; k0 += 4) {
            v2f b;
            b.x = wl[k0 * DOUT];
            b.y = wl[(k0 + 1) * DOUT];
            #pragma unroll
            for (int rt = 0; rt < 4; ++rt) {
                v2f a = *(const v2f*)(aM + rt * 16 * LDSTR + k0);
                acc[rt] = __builtin_amdgcn_wmma_f32_16x16x4_f32(
                    false, a, false, b, (short)0, acc[rt], false, false);
            }
        }
        // x @ Wr (accumulate)
        #pragma unroll 4
        for (int k0 = 0; k0 < FEAT; k0 += 4) {
            v2f b;
            b.x = wr[k0 * DOUT];
            b.y = wr[(k0 + 1) * DOUT];
            #pragma unroll
            for (int rt = 0; rt < 4; ++rt) {
                v2f a = *(const v2f*)(aX + rt * 16 * LDSTR + k0);
                acc[rt] = __builtin_amdgcn_wmma_f32_16x16x4_f32(
                    false, a, false, b, (short)0, acc[rt], false, false);
            }
        }

        // epilogue: bias + relu + guarded store
        const float bias = bl[nl];
        const bool  nv   = (n < DOUT);
        #pragma unroll
        for (int rt = 0; rt < 4; ++rt) {
            #pragma unroll
            for (int r = 0; r < 8; ++r) {
                int m = row0 + rt * 16 + r + moff;   // D: VGPR r -> M=r (+8 hi lanes)
                float v = acc[rt][r] + bias;
                if (RELU) v = fmaxf(v, 0.0f);
                if (nv && m < nrows) out[(long)m * DOUT + n] = v;
            }
        }
    }
}

// ---------------- host orchestration ----------------
extern "C" void kernel_launch(void* const* d_in, const int* in_sizes, int n_in,
                              void* d_out, int out_size, void* d_ws, size_t ws_size,
                              hipStream_t stream) {
    (void)in_sizes; (void)n_in; (void)out_size; (void)ws_size;

    const float* x    = (const float*)d_in[0];
    const int*   eidx = (const int*)d_in[1];   // [2,E] flat: first E = src, next E = dst
    const float* Wl0 = (const float*)d_in[3];
    const float* bl0 = (const float*)d_in[4];
    const float* Wr0 = (const float*)d_in[5];
    const float* Wl1 = (const float*)d_in[6];
    const float* bl1 = (const float*)d_in[7];
    const float* Wr1 = (const float*)d_in[8];
    const float* Wl2 = (const float*)d_in[9];
    const float* bl2 = (const float*)d_in[10];
    const float* Wr2 = (const float*)d_in[11];

    const int* src = eidx;
    const int* dst = eidx + NEDGES;

    float* ws     = (float*)d_ws;
    float* summed = ws;                              // N*128
    float* deg    = summed + (long)NNODES * FEAT;    // N
    float* h1     = deg + NNODES;                    // N*128
    float* h2     = h1 + (long)NNODES * FEAT;        // N*128
    float* outp   = (float*)d_out;                   // N*47

    const int nfeat4 = NNODES * (FEAT / 4);          // float4 count
    const int ndeg4  = NNODES / 4;
    const int ZB     = 256;
    const int gZeroF = (nfeat4 + ZB - 1) / ZB;
    const int gZeroD = (ndeg4 + ZB - 1) / ZB;
    const int gEdge  = (NEDGES + ZB - 1) / ZB;
    const int gScat  = (int)(((long)NEDGES * 32 + ZB - 1) / ZB);
    const int gRows  = (NNODES + 63) / 64;

    // degrees (once)
    zero_f32v4<<<gZeroD, ZB, 0, stream>>>((float4*)deg, ndeg4);
    degree_kernel<<<gEdge, ZB, 0, stream>>>(dst, deg, NEDGES);

    // layer 0: 128 -> 128, relu
    zero_f32v4<<<gZeroF, ZB, 0, stream>>>((float4*)summed, nfeat4);
    scatter_kernel<<<gScat, ZB, 0, stream>>>(x, src, dst, summed, NEDGES);
    sage_wmma<128, true><<<gRows, 256, 0, stream>>>(x, summed, deg, Wl0, bl0, Wr0,
                                                    h1, NNODES);

    // layer 1: 128 -> 128, relu
    zero_f32v4<<<gZeroF, ZB, 0, stream>>>((float4*)summed, nfeat4);
    scatter_kernel<<<gScat, ZB, 0, stream>>>(h1, src, dst, summed, NEDGES);
    sage_wmma<128, true><<<gRows, 256, 0, stream>>>(h1, summed, deg, Wl1, bl1, Wr1,
                                                    h2, NNODES);

    // layer 2: 128 -> 47, no relu
    zero_f32v4<<<gZeroF, ZB, 0, stream>>>((float4*)summed, nfeat4);
    scatter_kernel<<<gScat, ZB, 0, stream>>>(h2, src, dst, summed, NEDGES);
    sage_wmma<47, false><<<gRows, 256, 0, stream>>>(h2, summed, deg, Wl2, bl2, Wr2,
                                                    outp, NNODES);
}